// NSparseAttention_38757784879657
// MI455X (gfx1250) — compile-verified
//
#include <hip/hip_runtime.h>
#include <math.h>

// Problem constants (from reference)
#define BATCH 4
#define SEQ   2880
#define EDIM  512
#define BLK   64
#define NBLK  (SEQ / BLK)           // 45
#define CHUNK 128                   // E-dim tile held in LDS
#define QS_STRIDE 132               // 128 + 4 pad -> 4-bank skew per row
#define PS_STRIDE 68                // 64 + 4 pad

typedef __attribute__((ext_vector_type(2))) float v2f;
typedef __attribute__((ext_vector_type(8))) float v8f;

// ---------------------------------------------------------------------------
// Kernel 1: compute row-selection bitmask from phi[0:64].
// sel = union over columns j of top-8 indices i maximizing p[i]*p[j]
// (stable: strictly-greater keeps the lowest index on ties, like lax.top_k).
// ---------------------------------------------------------------------------
__global__ void nsattn_sel_kernel(const float* __restrict__ phi,
                                  unsigned long long* __restrict__ selmask) {
    __shared__ float p[BLK];
    __shared__ unsigned long long sm;
    const int t = threadIdx.x;
    if (t == 0) sm = 0ULL;
    if (t < BLK) p[t] = phi[t];
    __syncthreads();
    if (t < BLK) {
        const float pj = p[t];
        unsigned long long picked = 0ULL;
        for (int k = 0; k < 8; ++k) {
            float best = 0.0f;
            int bi = -1;
            for (int i = 0; i < BLK; ++i) {
                if (picked & (1ULL << i)) continue;
                float v = p[i] * pj;
                if (bi < 0 || v > best) { best = v; bi = i; }
            }
            picked |= (1ULL << bi);
        }
        atomicOr(&sm, picked);
    }
    __syncthreads();
    if (t == 0) *selmask = sm;
}

// ---------------------------------------------------------------------------
// Kernel 2: one workgroup (8 wave32) per (batch, 64-row block).
//   phase 1: scores = (x+pos+phi)(x+pos+phi)^T / sqrt(E)   via v_wmma f32 16x16x4
//   phase 2: row softmax over 64 cols; unselected rows -> probs row = 0
//   phase 3: out = probs @ x                               via v_wmma f32 16x16x4
// ---------------------------------------------------------------------------
__launch_bounds__(256)
__global__ void nsattn_main_kernel(const float* __restrict__ x,
                                   const float* __restrict__ phi,
                                   const float* __restrict__ pos,
                                   const unsigned long long* __restrict__ selmask,
                                   float* __restrict__ out) {
    __shared__ float qs[BLK][QS_STRIDE];   // q chunk (phase 1) / x chunk (phase 3)
    __shared__ float ps[BLK][PS_STRIDE];   // scores -> probs

    const int tid  = threadIdx.x;
    const int lane = tid & 31;
    const int wave = tid >> 5;             // 0..7
    const int half = lane >> 4;            // 0: lanes 0-15, 1: lanes 16-31
    const int l    = lane & 15;

    const int b  = blockIdx.x / NBLK;
    const int sb = blockIdx.x % NBLK;
    const long s0 = (long)sb * BLK;

    const float scale = 0.044194173824159216f;   // 1/sqrt(512)

    // -------- Phase 1: scores = q q^T, tiled over E in chunks of 128 --------
    v8f acc[2];
    for (int t = 0; t < 2; ++t)
        for (int i = 0; i < 8; ++i) acc[t][i] = 0.0f;

    for (int e0 = 0; e0 < EDIM; e0 += CHUNK) {
        // cooperative load: q chunk = x + pos_emb + phi  (64 x 128 f32)
        for (int i = 0; i < 8; ++i) {
            const int f   = i * 256 + tid;        // float4 index within chunk
            const int row = f >> 5;               // 32 float4 per row
            const int c4  = (f & 31) << 2;        // float col within chunk
            const float4 xv = *(const float4*)(x   + ((long)b * SEQ + s0 + row) * EDIM + e0 + c4);
            const float4 pv = *(const float4*)(pos + (s0 + row) * (long)EDIM + e0 + c4);
            const float4 fv = *(const float4*)(phi + e0 + c4);
            qs[row][c4 + 0] = xv.x + pv.x + fv.x;
            qs[row][c4 + 1] = xv.y + pv.y + fv.y;
            qs[row][c4 + 2] = xv.z + pv.z + fv.z;
            qs[row][c4 + 3] = xv.w + pv.w + fv.w;
        }
        __syncthreads();

        for (int t = 0; t < 2; ++t) {
            const int tile = wave * 2 + t;        // 16 score tiles, 2 per wave
            const int ti = tile >> 2, tj = tile & 3;
            const float* arow = &qs[ti * 16 + l][0];  // A: rows of q
            const float* brow = &qs[tj * 16 + l][0];  // B = q^T: cols are q rows
            for (int k0 = 0; k0 < CHUNK; k0 += 4) {
                v2f a, bb;
                // A 16x4: lanes 0-15 -> K=0,1 ; lanes 16-31 -> K=2,3 (ISA 7.12.2)
                a[0]  = arow[k0 + 2 * half];
                a[1]  = arow[k0 + 2 * half + 1];
                // B 4x16 row-striped: vgpr0 K=0(|2), vgpr1 K=1(|3), N = lane%16
                bb[0] = brow[k0 + 2 * half];
                bb[1] = brow[k0 + 2 * half + 1];
                acc[t] = __builtin_amdgcn_wmma_f32_16x16x4_f32(
                    false, a, false, bb, (short)0, acc[t], false, false);
            }
        }
        __syncthreads();   // protect qs before next chunk overwrite
    }

    // write scaled scores to LDS
    for (int t = 0; t < 2; ++t) {
        const int tile = wave * 2 + t;
        const int ti = tile >> 2, tj = tile & 3;
        for (int r = 0; r < 8; ++r)
            ps[ti * 16 + r + 8 * half][tj * 16 + l] = acc[t][r] * scale;
    }
    __syncthreads();

    // -------- Phase 2: per-row softmax; zero unselected rows --------
    const unsigned long long sm = *selmask;
    if (tid < BLK) {
        const int row = tid;
        if (!((sm >> row) & 1ULL)) {
            for (int j = 0; j < BLK; ++j) ps[row][j] = 0.0f;
        } else {
            float m = ps[row][0];
            for (int j = 1; j < BLK; ++j) m = fmaxf(m, ps[row][j]);
            float s = 0.0f;
            for (int j = 0; j < BLK; ++j) {
                float e = expf(ps[row][j] - m);
                ps[row][j] = e;
                s += e;
            }
            const float inv = 1.0f / s;
            for (int j = 0; j < BLK; ++j) ps[row][j] *= inv;
        }
    }
    __syncthreads();

    // -------- Phase 3: out = probs @ x, tiled over E in chunks of 128 --------
    for (int e0 = 0; e0 < EDIM; e0 += CHUNK) {
        // reload x chunk (L2-resident) into qs
        for (int i = 0; i < 8; ++i) {
            const int f   = i * 256 + tid;
            const int row = f >> 5;
            const int c4  = (f & 31) << 2;
            const float4 xv = *(const float4*)(x + ((long)b * SEQ + s0 + row) * EDIM + e0 + c4);
            qs[row][c4 + 0] = xv.x;
            qs[row][c4 + 1] = xv.y;
            qs[row][c4 + 2] = xv.z;
            qs[row][c4 + 3] = xv.w;
        }
        __syncthreads();

        for (int t = 0; t < 4; ++t) {
            const int tile = t * 8 + wave;        // 32 output tiles this chunk
            const int mt = tile >> 3, nt = tile & 7;
            v8f c;
            for (int i = 0; i < 8; ++i) c[i] = 0.0f;
            const float* arow = &ps[mt * 16 + l][0];
            for (int k0 = 0; k0 < BLK; k0 += 4) {
                v2f a, bb;
                a[0]  = arow[k0 + 2 * half];
                a[1]  = arow[k0 + 2 * half + 1];
                bb[0] = qs[k0 + 2 * half][nt * 16 + l];
                bb[1] = qs[k0 + 2 * half + 1][nt * 16 + l];
                c = __builtin_amdgcn_wmma_f32_16x16x4_f32(
                    false, a, false, bb, (short)0, c, false, false);
            }
            // C/D 16x16 f32: vgpr r -> M = r + 8*half, N = lane%16
            for (int r = 0; r < 8; ++r) {
                const long row = s0 + mt * 16 + r + 8 * half;
                out[((long)b * SEQ + row) * EDIM + e0 + nt * 16 + l] = c[r];
            }
        }
        __syncthreads();
    }
}

// ---------------------------------------------------------------------------
extern "C" void kernel_launch(void* const* d_in, const int* in_sizes, int n_in,
                              void* d_out, int out_size, void* d_ws, size_t ws_size,
                              hipStream_t stream) {
    (void)in_sizes; (void)n_in; (void)out_size; (void)ws_size;
    const float* x   = (const float*)d_in[0];   // [B,S,E]
    const float* phi = (const float*)d_in[1];   // [1,1,E]
    const float* pos = (const float*)d_in[2];   // [S,E]
    float* out = (float*)d_out;                 // [B,S,E]
    unsigned long long* selmask = (unsigned long long*)d_ws;

    nsattn_sel_kernel<<<1, 64, 0, stream>>>(phi, selmask);
    nsattn_main_kernel<<<dim3(BATCH * NBLK), dim3(256), 0, stream>>>(
        x, phi, pos, selmask, out);
}